// augmentation_layer_46935402610702
// MI455X (gfx1250) — compile-verified
//
#include <hip/hip_runtime.h>

// Time-warp augmentation for MI455X (gfx1250, wave32).
//
// Roofline: 134 MB in + 134 MB out => ~11.5 us floor at 23.3 TB/s.
//  - closed-form not-a-knot moment solve M = G(6x4) @ rhs(4), 16 channels per
//    wave via V_WMMA_F32_16X16X4_F32 (D rows 0..5 = M0..M5).
//  - transpose x [B,T,C] -> xT [B,C,T] in workspace so per-channel time
//    gathers are contiguous.
//  - invert the monotone warped grid with Newton (v_rcp_f32) + running
//    pointer with register-resident interval coefficients: ~6 VALU + 1
//    contiguous load per output element. No IEEE div in the hot path.
//  - results staged in LDS tile, written back coalesced in [T,C] order.

#define BDIM 128
#define TDIM 4096
#define CDIM 64
#define HSP  819.0f      // knot spacing (T-1)/(n-1) = 4095/5
#define QPL  8           // queries per lane
#define QTILE (32*QPL)   // 256 queries per block tile
#define CTILE 32         // channels per block
#define CO_STRIDE 24     // 20 scaled coefs + S5 + xplast + pad

typedef __attribute__((ext_vector_type(2))) float v2f;
typedef __attribute__((ext_vector_type(8))) float v8f;

#define RCPF(x) __builtin_amdgcn_rcpf(x)

// G = columns 1..4 of inv(A) for the uniform not-a-knot system (n=6, h=819),
// padded to 16 rows for the 16x4 WMMA A operand (derived analytically).
__constant__ float G_CONST[16][4] = {
  { 17.0f/(45.0f*HSP), -4.0f/(15.0f*HSP),  1.0f/(15.0f*HSP), -1.0f/(90.0f*HSP)},
  {  1.0f/( 6.0f*HSP),  0.0f,              0.0f,              0.0f            },
  { -2.0f/(45.0f*HSP),  4.0f/(15.0f*HSP), -1.0f/(15.0f*HSP),  1.0f/(90.0f*HSP)},
  {  1.0f/(90.0f*HSP), -1.0f/(15.0f*HSP),  4.0f/(15.0f*HSP), -2.0f/(45.0f*HSP)},
  {  0.0f,              0.0f,              0.0f,              1.0f/( 6.0f*HSP)},
  { -1.0f/(90.0f*HSP),  1.0f/(15.0f*HSP), -4.0f/(15.0f*HSP), 17.0f/(45.0f*HSP)},
  {0,0,0,0},{0,0,0,0},{0,0,0,0},{0,0,0,0},{0,0,0,0},
  {0,0,0,0},{0,0,0,0},{0,0,0,0},{0,0,0,0},{0,0,0,0}
};

// xp[j] = clip(scaled_spline(j), 0, 4095); coefs in LDS are pre-scaled.
__device__ __forceinline__ float xp_eval(const float* co, int j) {
  int k = j / 819; if (k > 4) k = 4;
  float t = (float)(j - k * 819);
  const float* ck = co + 4 * k;
  float v = ck[0] + t * (ck[1] + t * (ck[2] + t * ck[3]));
  return fminf(fmaxf(v, 0.0f), 4095.0f);
}

__global__ __launch_bounds__(256)
void transpose_kernel(const float* __restrict__ x, float* __restrict__ xT) {
  __shared__ float tile[64][65];
  const int b  = blockIdx.y;
  const int t0 = blockIdx.x * 64;
  const int tid = threadIdx.x;
  const int cl = tid & 63, row = tid >> 6;
  const float* src = x + ((size_t)b * TDIM + t0) * CDIM;
  #pragma unroll
  for (int tt = row; tt < 64; tt += 4)
    tile[tt][cl] = src[(size_t)tt * CDIM + cl];            // coalesced in c
  __syncthreads();
  const int tl = tid & 63, cr = tid >> 6;
  float* dst = xT + (size_t)b * CDIM * TDIM + t0;
  #pragma unroll
  for (int cc = cr; cc < 64; cc += 4)
    dst[(size_t)cc * TDIM + tl] = tile[tl][cc];            // coalesced in t
}

template <int FAST>
__global__ __launch_bounds__(256)
void timewarp_kernel(const float* __restrict__ x, const float* __restrict__ xT,
                     const float* __restrict__ rwarp, float* __restrict__ out) {
  __shared__ float co_s[CTILE * CO_STRIDE];           // per-channel spline data
  __shared__ float ftile[CTILE * (QTILE + 1)];        // result tile (transposed)
  const int tid  = threadIdx.x;
  const int wave = tid >> 5, lane = tid & 31;
  const int c0   = blockIdx.y * CTILE;                // channel half
  const int b    = blockIdx.z;
  const int q0   = blockIdx.x * QTILE;

  // ---------------- spline setup: 2 waves x 16 channels, one WMMA each -----
  if (wave < 2) {
    const int col = lane & 15, half = lane >> 4;
    const int cl = wave * 16 + col;                   // local channel
    const int cg = c0 + cl;                           // global channel
    float y[6];
    #pragma unroll
    for (int k = 0; k < 6; ++k) {
      float r = rwarp[((size_t)b * 6 + k) * CDIM + cg];
      y[k] = (HSP * (float)k) * r;
    }
    float rhs[5];
    #pragma unroll
    for (int j = 1; j <= 4; ++j)
      rhs[j] = (y[j+1] - 2.0f*y[j] + y[j-1]) * (6.0f / HSP);
    // A (16x4 f32): lane row=col, VGPR0=K{0|2}, VGPR1=K{1|3} by lane half
    v2f av; av.x = G_CONST[col][half*2]; av.y = G_CONST[col][half*2+1];
    // B (4x16 f32): VGPR0 rows K{0|2}, VGPR1 rows K{1|3}, cols striped on lanes
    v2f bv; bv.x = half ? rhs[3] : rhs[1]; bv.y = half ? rhs[4] : rhs[2];
    v8f acc = {0.f,0.f,0.f,0.f,0.f,0.f,0.f,0.f};
    acc = __builtin_amdgcn_wmma_f32_16x16x4_f32(
        /*neg_a=*/false, av, /*neg_b=*/false, bv,
        /*c_mod=*/(short)0, acc, /*reuse_a=*/false, /*reuse_b=*/false);
    if (half == 0) {                    // lanes 0..15: rows 0..5 = M0..M5
      float M[6];
      #pragma unroll
      for (int k = 0; k < 6; ++k) M[k] = acc[k];
      float c1[5], hm[5], d3[5];
      #pragma unroll
      for (int k = 0; k < 5; ++k) {
        c1[k] = (y[k+1]-y[k]) * (1.0f/HSP) - (HSP/6.0f) * (2.0f*M[k] + M[k+1]);
        hm[k] = 0.5f * M[k];
        d3[k] = (M[k+1] - M[k]) * (1.0f / (6.0f * HSP));
      }
      float t = HSP;
      float tw_last = y[4] + t*(c1[4] + t*(hm[4] + t*d3[4]));
      float scale = 4095.0f / tw_last;               // once per channel: ok
      float* co = &co_s[cl * CO_STRIDE];
      #pragma unroll
      for (int k = 0; k < 5; ++k) {                  // pre-scaled coefficients
        co[4*k+0] = scale * y[k];
        co[4*k+1] = scale * c1[k];
        co[4*k+2] = scale * hm[k];
        co[4*k+3] = scale * d3[k];
      }
      co[20] = scale * y[5];                                   // S5
      co[21] = fminf(fmaxf(scale * tw_last, 0.0f), 4095.0f);   // xp[T-1]
    }
  }
  __syncthreads();

  const float* xb  = x  + (size_t)b * TDIM * CDIM;
  const float* xTb = xT + (size_t)b * CDIM * TDIM;

  // -------- main: wave owns a channel; lane owns QPL consecutive queries ----
  for (int cl = wave; cl < CTILE; cl += 8) {
    const int cg = c0 + cl;
    const float* co = &co_s[cl * CO_STRIDE];
    const float* fp = xTb + (size_t)cg * TDIM;
    const float xplast = co[21];
    const int   qf = q0 + lane * QPL;
    const float q  = (float)qf;

    // spline interval of the answer (S_k == co[4k]), then Newton on the cubic
    int k = 0;
    k += (q >= co[4]); k += (q >= co[8]); k += (q >= co[12]); k += (q >= co[16]);
    float sy = co[4*k], sc1 = co[4*k+1], shm = co[4*k+2], sd3 = co[4*k+3];
    const float Sk1 = (k < 4) ? co[4*k+4] : co[20];
    float den = Sk1 - sy;
    den = (fabsf(den) < 1e-20f) ? 1e-20f : den;
    float t = (q - sy) * HSP * RCPF(den);
    #pragma unroll
    for (int it = 0; it < 4; ++it) {
      float fv = sy + t*(sc1 + t*(shm + t*sd3)) - q;
      float dv = sc1 + t*(2.0f*shm + 3.0f*sd3*t);
      dv = (fabsf(dv) < 1e-12f) ? ((dv < 0.0f) ? -1e-12f : 1e-12f) : dv;
      t -= fv * RCPF(dv);
      t = fminf(fmaxf(t, -900.0f), 1700.0f);
    }
    int i = k * 819 + (int)floorf(t);
    i = (i < 1) ? 1 : ((i > 4095) ? 4095 : i);
    // fixup to exact searchsorted(...,'right') clipped to [1, T-1]
    int g = 0;
    while (i > 1    && xp_eval(co, i-1) >  q && g < 96) { --i; ++g; }
    g = 0;
    while (i < 4095 && xp_eval(co, i)   <= q && g < 96) { ++i; ++g; }

    // register-resident state for the running pointer
    k = i / 819; if (k > 4) k = 4;
    float tc = (float)(i - k * 819);
    sy = co[4*k]; sc1 = co[4*k+1]; shm = co[4*k+2]; sd3 = co[4*k+3];
    float cur   = fminf(fmaxf(sy + tc*(sc1 + tc*(shm + tc*sd3)), 0.0f), 4095.0f);
    float prev  = xp_eval(co, i - 1);
    float fcur  = FAST ? fp[i]   : xb[(size_t)i     * CDIM + cg];
    float fprev = FAST ? fp[i-1] : xb[(size_t)(i-1) * CDIM + cg];

    float qv = q;
    #pragma unroll 1
    for (int s = 0; s < QPL; ++s) {
      #pragma unroll 1
      while (i < 4095 && cur <= qv) {   // monotone running pointer
        ++i; tc += 1.0f;
        if (tc >= HSP && k < 4) {       // crossed a knot: reload interval coefs
          ++k; tc -= HSP;
          sy = co[4*k]; sc1 = co[4*k+1]; shm = co[4*k+2]; sd3 = co[4*k+3];
        }
        prev = cur; fprev = fcur;
        cur  = fminf(fmaxf(sy + tc*(sc1 + tc*(shm + tc*sd3)), 0.0f), 4095.0f);
        fcur = FAST ? fp[i] : xb[(size_t)i * CDIM + cg];
      }
      float dx = cur - prev;
      float f  = fprev + (qv - prev) * (fcur - fprev) * RCPF(dx);
      f = (dx == 0.0f) ? fcur : f;
      if (qv > xplast)
        f = FAST ? fp[TDIM-1] : xb[(size_t)(TDIM-1) * CDIM + cg];
      ftile[cl * (QTILE + 1) + (lane * QPL + s)] = f;
      qv += 1.0f;
    }
  }
  __syncthreads();

  // -------- coalesced writeback in [T, C] order ----------------------------
  const int cw = tid & 31, r0 = tid >> 5;
  float* ob = out + ((size_t)b * TDIM + q0) * CDIM + c0;
  #pragma unroll
  for (int qq = r0; qq < QTILE; qq += 8)
    ob[(size_t)qq * CDIM + cw] = ftile[cw * (QTILE + 1) + qq];
}

extern "C" void kernel_launch(void* const* d_in, const int* in_sizes, int n_in,
                              void* d_out, int out_size, void* d_ws, size_t ws_size,
                              hipStream_t stream) {
  (void)in_sizes; (void)n_in; (void)out_size;
  const float* x  = (const float*)d_in[0];   // [B,1,T,C] f32
  const float* rw = (const float*)d_in[1];   // [B,6,C]   f32
  float* out = (float*)d_out;                // [B,1,T,C] f32
  float* xT  = (float*)d_ws;

  const size_t need = (size_t)BDIM * TDIM * CDIM * sizeof(float);
  const bool fast = (ws_size >= need) && (d_ws != nullptr);

  if (fast) {
    transpose_kernel<<<dim3(TDIM / 64, BDIM), 256, 0, stream>>>(x, xT);
    timewarp_kernel<1><<<dim3(TDIM / QTILE, CDIM / CTILE, BDIM), 256, 0, stream>>>(
        x, xT, rw, out);
  } else {
    timewarp_kernel<0><<<dim3(TDIM / QTILE, CDIM / CTILE, BDIM), 256, 0, stream>>>(
        x, nullptr, rw, out);
  }
}